// Deep_GATE_Conv_66340064854190
// MI455X (gfx1250) — compile-verified
//
#include <hip/hip_runtime.h>

// Deep GATE Conv (GAT attention + PNA aggregation) for MI455X / gfx1250.
// All dense GEMMs run on V_WMMA_F32_16X16X4_F32 (fp32 matrix pipe, matches
// reference precision). Irregular scatter uses fp32 global atomics + ordered
// uint encoding for float atomic min/max.

#define N_NODES 50000
#define E_EDGES 800000
#define HH 2
#define CC 64
#define CEE 32
#define NODE_IN 128
#define EDGE_IN 32
#define NEG_SLOPE 0.2f

typedef __attribute__((ext_vector_type(2))) float v2f;
typedef __attribute__((ext_vector_type(8))) float v8f;

// ---- order-preserving float <-> uint encoding (for atomic min/max/max) ----
__device__ __forceinline__ unsigned fenc(float f) {
  unsigned u = __float_as_uint(f);
  return (u & 0x80000000u) ? ~u : (u | 0x80000000u);
}
__device__ __forceinline__ float fdec(unsigned u) {
  unsigned v = (u & 0x80000000u) ? (u & 0x7FFFFFFFu) : ~u;
  return __uint_as_float(v);
}
__device__ __forceinline__ float lrelu(float v) { return v > 0.f ? v : NEG_SLOPE * v; }

// D = A(16x4,f32) * B(4x16,f32) + C(16x16,f32)
__device__ __forceinline__ v8f wmma_f32(v2f a, v2f b, v8f c) {
  return __builtin_amdgcn_wmma_f32_16x16x4_f32(false, a, false, b, (short)0, c,
                                               false, false);
}

// ---------------------------------------------------------------- init ----
__global__ void k_init(float* denom, float* cnt, float* sum, float* sumsq,
                       unsigned* segmax, unsigned* minenc, unsigned* maxenc,
                       float* avglog) {
  int stride = gridDim.x * blockDim.x;
  int tid = blockIdx.x * blockDim.x + threadIdx.x;
  for (int i = tid; i < N_NODES * NODE_IN; i += stride) {
    sum[i] = 0.f;
    sumsq[i] = 0.f;
    minenc[i] = 0xFFFFFFFFu;  // decodes above any real value for atomic-min
    maxenc[i] = 0u;           // decodes below any real value for atomic-max
  }
  for (int i = tid; i < N_NODES * HH; i += stride) {
    denom[i] = 0.f;
    segmax[i] = 0u;
  }
  for (int i = tid; i < N_NODES; i += stride) cnt[i] = 0.f;
  if (tid == 0) avglog[0] = 0.f;
}

// ------------------------------------------- xl = x @ W_l + b_l (WMMA) ----
// block = 8 waves; each wave computes one 16x16 tile of a 16x128 row-band.
__global__ void __launch_bounds__(256) k_node_lin(const float* __restrict__ x,
                                                  const float* __restrict__ Wl,
                                                  const float* __restrict__ bl,
                                                  float* __restrict__ xl) {
  __shared__ float sX[16][NODE_IN + 4];  // pad: column reads hit distinct banks
  int tid = threadIdx.x;
  int row0 = blockIdx.x * 16;  // 3125 * 16 == 50000 exactly
  for (int i = tid; i < 16 * NODE_IN; i += 256) {
    int r = i >> 7, c = i & 127;
    sX[r][c] = x[(size_t)(row0 + r) * NODE_IN + c];
  }
  __syncthreads();
  int wave = tid >> 5, lane = tid & 31;
  int n0 = wave * 16;
  int lrow = lane & 15;
  int kh = (lane >> 4) << 1;  // lanes 16-31 carry K+2,K+3
  int mh = (lane >> 4) << 3;  // C/D: lanes 16-31 carry rows M+8
  v8f acc = {};
  for (int k = 0; k < NODE_IN; k += 4) {
    v2f a, b;
    a.x = sX[lrow][k + kh];
    a.y = sX[lrow][k + kh + 1];
    b.x = Wl[(size_t)(k + kh) * 128 + n0 + lrow];      // W_l L2-resident (64KB)
    b.y = Wl[(size_t)(k + kh + 1) * 128 + n0 + lrow];
    acc = wmma_f32(a, b, acc);
  }
  float bias = bl[n0 + lrow];
  for (int r = 0; r < 8; ++r)
    xl[(size_t)(row0 + r + mh) * 128 + n0 + lrow] = acc[r] + bias;
}

// --------------- e = edge_attr @ W_e + b_e ; edge_out = e @ W_eo + b_eo ----
// block = 32 edges; GEMM1 (K=32) with all 8 waves, tile kept in LDS, then
// GEMM2 (K=64) with 4 waves straight into d_out.
__global__ void __launch_bounds__(256) k_edge_lin(
    const float* __restrict__ ea, const float* __restrict__ We,
    const float* __restrict__ be, const float* __restrict__ Weo,
    const float* __restrict__ beo, float* __restrict__ e_out,
    float* __restrict__ edge_out) {
  __shared__ float sA[32][EDGE_IN + 1];
  __shared__ float sWe[EDGE_IN][64];
  __shared__ float sE[32][65];
  __shared__ float sWeo[64][32];
  int tid = threadIdx.x;
  int row0 = blockIdx.x * 32;  // 25000 * 32 == 800000 exactly
  for (int i = tid; i < 32 * 32; i += 256) {
    int r = i >> 5, c = i & 31;
    sA[r][c] = ea[(size_t)(row0 + r) * 32 + c];
  }
  for (int i = tid; i < 32 * 64; i += 256) sWe[i >> 6][i & 63] = We[i];
  for (int i = tid; i < 64 * 32; i += 256) sWeo[i >> 5][i & 31] = Weo[i];
  __syncthreads();
  int wave = tid >> 5, lane = tid & 31;
  int lrow = lane & 15, kh = (lane >> 4) << 1, mh = (lane >> 4) << 3;
  int rt = wave >> 2, ct = wave & 3;
  v8f acc = {};
  for (int k = 0; k < 32; k += 4) {
    v2f a, b;
    a.x = sA[rt * 16 + lrow][k + kh];
    a.y = sA[rt * 16 + lrow][k + kh + 1];
    b.x = sWe[k + kh][ct * 16 + lrow];
    b.y = sWe[k + kh + 1][ct * 16 + lrow];
    acc = wmma_f32(a, b, acc);
  }
  float bias = be[ct * 16 + lrow];
  for (int r = 0; r < 8; ++r) {
    int m = rt * 16 + r + mh;
    float v = acc[r] + bias;
    sE[m][ct * 16 + lrow] = v;
    e_out[(size_t)(row0 + m) * 64 + ct * 16 + lrow] = v;
  }
  __syncthreads();
  if (wave < 4) {
    int rt2 = wave >> 1, ct2 = wave & 1;
    v8f a2 = {};
    for (int k = 0; k < 64; k += 4) {
      v2f a, b;
      a.x = sE[rt2 * 16 + lrow][k + kh];
      a.y = sE[rt2 * 16 + lrow][k + kh + 1];
      b.x = sWeo[k + kh][ct2 * 16 + lrow];
      b.y = sWeo[k + kh + 1][ct2 * 16 + lrow];
      a2 = wmma_f32(a, b, a2);
    }
    float b2 = beo[ct2 * 16 + lrow];
    for (int r = 0; r < 8; ++r) {
      int m = rt2 * 16 + r + mh;
      edge_out[(size_t)(row0 + m) * 32 + ct2 * 16 + lrow] = a2[r] + b2;
    }
  }
}

// -------- per-edge attention logits + atomic segment-max + degree count ----
// one wave (32 lanes) per edge: coalesced 128f gathers, shuffle reduction.
__global__ void __launch_bounds__(256) k_logits(
    const float* __restrict__ xl, const float* __restrict__ ew,
    const int* __restrict__ ei, const float* __restrict__ att,
    float* __restrict__ logit, unsigned* __restrict__ segmax,
    float* __restrict__ cnt) {
  int wave = threadIdx.x >> 5, lane = threadIdx.x & 31;
  int e = blockIdx.x * 8 + wave;  // 100000 * 8 == 800000 exactly
  int src = ei[e], dst = ei[E_EDGES + e];
  const float* xi = xl + (size_t)dst * 128;
  const float* xj = xl + (size_t)src * 128;
  float p0 = 0.f, p1 = 0.f;
  for (int t = lane; t < 128; t += 32) {
    int h = t >> 6, c = t & 63;
    float v = lrelu(xi[t]) * att[h * 160 + c] +        // x_i slot
              lrelu(xj[t]) * att[h * 160 + 64 + c];    // x_j slot
    if (h) p1 += v; else p0 += v;
  }
  const float* ee = ew + (size_t)e * 64;
  for (int t = lane; t < 64; t += 32) {
    int h = t >> 5, c = t & 31;
    float v = lrelu(ee[t]) * att[h * 160 + 128 + c];   // edge slot
    if (h) p1 += v; else p0 += v;
  }
  for (int off = 16; off; off >>= 1) {
    p0 += __shfl_xor(p0, off, 32);
    p1 += __shfl_xor(p1, off, 32);
  }
  if (lane == 0) {
    logit[(size_t)e * 2] = p0;
    logit[(size_t)e * 2 + 1] = p1;
    atomicMax(&segmax[dst * 2], fenc(p0));
    atomicMax(&segmax[dst * 2 + 1], fenc(p1));
    atomicAdd(&cnt[dst], 1.f);
  }
}

// ---------------------- ex = exp(logit - segmax[dst]); denom += ex ---------
__global__ void k_exp(float* __restrict__ logit,
                      const unsigned* __restrict__ segmax,
                      const int* __restrict__ ei, float* __restrict__ denom) {
  int i = blockIdx.x * blockDim.x + threadIdx.x;  // E*H = 1.6M exact
  int e = i >> 1, h = i & 1;
  int dst = ei[E_EDGES + e];
  float ex = expf(logit[i] - fdec(segmax[dst * 2 + h]));
  logit[i] = ex;  // buffer now holds ex
  atomicAdd(&denom[dst * 2 + h], ex);
}

// ----------------------- avg_log = mean_n log(cnt[n]+1) (block reduce) ----
__global__ void __launch_bounds__(256) k_avglog(const float* __restrict__ cnt,
                                                float* __restrict__ avglog) {
  __shared__ float red[256];
  int i = blockIdx.x * 256 + threadIdx.x;
  red[threadIdx.x] = (i < N_NODES) ? logf(cnt[i] + 1.f) : 0.f;
  __syncthreads();
  for (int s = 128; s; s >>= 1) {
    if (threadIdx.x < s) red[threadIdx.x] += red[threadIdx.x + s];
    __syncthreads();
  }
  if (threadIdx.x == 0) atomicAdd(avglog, red[0]);
}

// --------------- PNA scatter: sum / sumsq / min / max of alpha * x_j -------
__global__ void __launch_bounds__(256) k_aggregate(
    const float* __restrict__ xl, const float* __restrict__ ex,
    const float* __restrict__ denom, const int* __restrict__ ei,
    float* __restrict__ sum, float* __restrict__ sumsq,
    unsigned* __restrict__ minenc, unsigned* __restrict__ maxenc) {
  int wave = threadIdx.x >> 5, lane = threadIdx.x & 31;
  int e = blockIdx.x * 8 + wave;
  int src = ei[e], dst = ei[E_EDGES + e];
  float a0 = ex[(size_t)e * 2] / (denom[dst * 2] + 1e-16f);
  float a1 = ex[(size_t)e * 2 + 1] / (denom[dst * 2 + 1] + 1e-16f);
  const float* xj = xl + (size_t)src * 128;
  for (int t = lane; t < 128; t += 32) {
    float msg = xj[t] * ((t & 64) ? a1 : a0);
    int idx = dst * 128 + t;
    atomicAdd(&sum[idx], msg);
    atomicAdd(&sumsq[idx], msg * msg);
    atomicMax(&maxenc[idx], fenc(msg));
    atomicMin(&minenc[idx], fenc(msg));
  }
}

// ------- finalize PNA features + node_out = agg(1536) @ W_no (WMMA) -------
// out_att == out (amp * 1/amp cancels), so agg = [out, out*s, out] per head.
// Build agg in 256-wide K-chunks in LDS; 8 waves = 4 N-tiles x 2 K-halves.
__global__ void __launch_bounds__(256) k_finalize(
    const float* __restrict__ sum, const float* __restrict__ sumsq,
    const unsigned* __restrict__ minenc, const unsigned* __restrict__ maxenc,
    const float* __restrict__ cnt, const float* __restrict__ avglog,
    const float* __restrict__ Wno, const float* __restrict__ bno,
    float* __restrict__ node_out) {
  __shared__ float sChunk[16][257];  // 16 nodes x 256 K, padded
  __shared__ float sPart[4 * 256];
  int tid = threadIdx.x;
  int row0 = blockIdx.x * 16;
  int wave = tid >> 5, lane = tid & 31;
  int lrow = lane & 15, kh = (lane >> 4) << 1, mh = (lane >> 4) << 3;
  int ct = wave & 3, n0 = ct * 16;
  int ks0 = (wave >> 2) * 128;  // K-half within the chunk
  float avg = avglog[0] / (float)N_NODES;
  v8f acc = {};
  for (int g = 0; g < 6; ++g) {  // chunk = (head h, section {out,amp,att})
    int h = g / 3, sec = g % 3;
    int kbase = h * 768 + sec * 256;
    __syncthreads();  // previous chunk fully consumed
    for (int i = tid; i < 16 * 256; i += 256) {
      int m = i >> 8, j = i & 255;
      int ag = j >> 6, c = j & 63;  // aggregator: mean|min|max|std
      int node = row0 + m;
      float cn = cnt[node];
      float cc = fmaxf(cn, 1.f);
      int idx = node * 128 + h * 64 + c;
      float mean = sum[idx] / cc;
      float val;
      if (ag == 0) val = mean;
      else if (ag == 1) val = (cn == 0.f) ? 0.f : fdec(minenc[idx]);
      else if (ag == 2) val = (cn == 0.f) ? 0.f : fdec(maxenc[idx]);
      else {
        float var = sumsq[idx] / cc - mean * mean;
        val = sqrtf(fmaxf(var, 0.f) + 1e-5f);
      }
      if (sec == 1) val *= logf(cc + 1.f) / avg;  // amplification scaler
      sChunk[m][j] = val;
    }
    __syncthreads();
    const float* Bp = Wno + (size_t)kbase * 64;
    for (int k = ks0; k < ks0 + 128; k += 4) {
      v2f a, b;
      a.x = sChunk[lrow][k + kh];
      a.y = sChunk[lrow][k + kh + 1];
      b.x = Bp[(size_t)(k + kh) * 64 + n0 + lrow];
      b.y = Bp[(size_t)(k + kh + 1) * 64 + n0 + lrow];
      acc = wmma_f32(a, b, acc);
    }
  }
  __syncthreads();
  if (wave >= 4)
    for (int r = 0; r < 8; ++r) sPart[ct * 256 + r * 32 + lane] = acc[r];
  __syncthreads();
  if (wave < 4) {
    float bias = bno[n0 + lrow];
    for (int r = 0; r < 8; ++r) {
      float v = acc[r] + sPart[ct * 256 + r * 32 + lane] + bias;
      node_out[(size_t)(row0 + r + mh) * 64 + n0 + lrow] = v;
    }
  }
}

// --------------------------------------------------------------- launch ---
extern "C" void kernel_launch(void* const* d_in, const int* in_sizes, int n_in,
                              void* d_out, int out_size, void* d_ws,
                              size_t ws_size, hipStream_t stream) {
  const float* x = (const float*)d_in[0];
  const float* ea = (const float*)d_in[1];
  const int* ei = (const int*)d_in[2];
  const float* Wl = (const float*)d_in[3];
  const float* bl = (const float*)d_in[4];
  const float* We = (const float*)d_in[5];
  const float* be = (const float*)d_in[6];
  const float* att = (const float*)d_in[7];
  const float* Wno = (const float*)d_in[8];
  const float* bno = (const float*)d_in[9];
  const float* Weo = (const float*)d_in[10];
  const float* beo = (const float*)d_in[11];
  float* node_out = (float*)d_out;                       // [N, 64]
  float* edge_out = node_out + (size_t)N_NODES * CC;     // [E, 32]

  // workspace layout (~340 MB)
  float* xl_ws = (float*)d_ws;                               // [N,128]
  float* e_ws = xl_ws + (size_t)N_NODES * 128;               // [E,64]
  float* logit_ws = e_ws + (size_t)E_EDGES * 64;             // [E,2] logit->ex
  float* denom_ws = logit_ws + (size_t)E_EDGES * 2;          // [N,2]
  float* cnt_ws = denom_ws + (size_t)N_NODES * 2;            // [N]
  float* sum_ws = cnt_ws + N_NODES;                          // [N,128]
  float* sumsq_ws = sum_ws + (size_t)N_NODES * 128;          // [N,128]
  unsigned* segmax_ws = (unsigned*)(sumsq_ws + (size_t)N_NODES * 128);
  unsigned* minenc_ws = segmax_ws + (size_t)N_NODES * 2;     // [N,128]
  unsigned* maxenc_ws = minenc_ws + (size_t)N_NODES * 128;   // [N,128]
  float* avglog_ws = (float*)(maxenc_ws + (size_t)N_NODES * 128);

  k_init<<<4096, 256, 0, stream>>>(denom_ws, cnt_ws, sum_ws, sumsq_ws,
                                   segmax_ws, minenc_ws, maxenc_ws, avglog_ws);
  k_node_lin<<<N_NODES / 16, 256, 0, stream>>>(x, Wl, bl, xl_ws);
  k_edge_lin<<<E_EDGES / 32, 256, 0, stream>>>(ea, We, be, Weo, beo, e_ws,
                                               edge_out);
  k_logits<<<E_EDGES / 8, 256, 0, stream>>>(xl_ws, e_ws, ei, att, logit_ws,
                                            segmax_ws, cnt_ws);
  k_exp<<<(E_EDGES * HH) / 256, 256, 0, stream>>>(logit_ws, segmax_ws, ei,
                                                  denom_ws);
  k_avglog<<<(N_NODES + 255) / 256, 256, 0, stream>>>(cnt_ws, avglog_ws);
  k_aggregate<<<E_EDGES / 8, 256, 0, stream>>>(xl_ws, logit_ws, denom_ws, ei,
                                               sum_ws, sumsq_ws, minenc_ws,
                                               maxenc_ws);
  k_finalize<<<N_NODES / 16, 256, 0, stream>>>(sum_ws, sumsq_ws, minenc_ws,
                                               maxenc_ws, cnt_ws, avglog_ws,
                                               Wno, bno, node_out);
}